// ConvCapsule_26989574488807
// MI455X (gfx1250) — compile-verified
//
#include <hip/hip_runtime.h>
#include <math.h>

// Problem constants (from reference)
#define IC   32
#define IND  16
#define OC   32
#define OD   16
#define KK   3
#define STR  2
#define NB   8
#define H    14
#define WD   14
#define OHH  6
#define OWW  6
#define OHW  36                 // 6*6
#define G    288                // K*K*IC
#define PP   512                // OC*OD
#define XCH  (IC*(IND+1))       // 544
#define NBHW (NB*OHW)           // 288
#define NBOHW (NB*OC*OHW)       // 9216
#define NTILES (NBHW/16)        // 18

typedef float v2f __attribute__((ext_vector_type(2)));
typedef float v8f __attribute__((ext_vector_type(8)));

// ---------------------------------------------------------------------------
// Kernel 1: pack vector patches (im2col, physical g decode) + activation
// patches (relabeled g decode).  vp[bhw][g][ind], ap[bhw][g]
// ---------------------------------------------------------------------------
__global__ void pack_kernel(const float* __restrict__ x,
                            float* __restrict__ vp, float* __restrict__ ap) {
    int t = blockIdx.x * blockDim.x + threadIdx.x;
    if (t >= NBHW * G) return;
    int bhw = t / G, g = t % G;
    int b = bhw / OHW, hw = bhw % OHW;
    int h = hw / OWW, w = hw % OWW;

    // physical decode: g = (ki*K + kj)*IC + cc
    int ki = g / (KK * IC);
    int kj = (g / IC) % KK;
    int cc = g % IC;
    int xr = h * STR + ki, xc = w * STR + kj;
    const float* xb = x + (size_t)b * XCH * (H * WD) + xr * WD + xc;
    float* vpo = vp + (size_t)t * IND;
#pragma unroll
    for (int d = 0; d < IND; ++d)
        vpo[d] = xb[(size_t)(cc * (IND + 1) + 1 + d) * (H * WD)];

    // relabeled decode: g = (c*K + i)*K + j  (activations)
    int c2 = g / (KK * KK);
    int i2 = (g / KK) % KK;
    int j2 = g % KK;
    ap[t] = x[((size_t)b * XCH + c2 * (IND + 1)) * (H * WD) +
              (h * STR + i2) * WD + (w * STR + j2)];
}

// ---------------------------------------------------------------------------
// Kernel 2: votes via V_WMMA_F32_16X16X4_F32.
// One wave per (g, o): A (W tile) + bias C-init held in registers, loop over
// the 18 N-tiles (16 bhw sites each); K=16 via 4 chained WMMAs.
// votes layout: [bohw][g][d]
// ---------------------------------------------------------------------------
__global__ __launch_bounds__(32)
void votes_kernel(const float* __restrict__ Wm,
                  const float* __restrict__ bias,
                  const float* __restrict__ vp,
                  float* __restrict__ votes) {
    int o = blockIdx.x;             // 0..31  p tile (p = o*16 + d)
    int g = blockIdx.y;             // 0..287
    int lane = threadIdx.x;         // wave32
    int half = lane >> 4;
    int lh   = lane & 15;

    // A: W[g][p][k]; lane M = lh (p row), K pair selected by lane half.
    const float* Arow = Wm + ((size_t)g * PP + o * OD + lh) * IND + half * 2;
    v2f a[4];
#pragma unroll
    for (int kc = 0; kc < 4; ++kc) {
        a[kc].x = Arow[kc * 4 + 0];
        a[kc].y = Arow[kc * 4 + 1];
    }

    // C init from bias: lane row r holds d = half*8 + r
    const float* bsrc = bias + (size_t)g * PP + o * OD + half * 8;
    v8f cb;
#pragma unroll
    for (int r = 0; r < 8; ++r) cb[r] = bsrc[r];

    for (int nt = 0; nt < NTILES; ++nt) {
        int bhw = nt * 16 + lh;
        const float* Brow = vp + ((size_t)bhw * G + g) * IND + half * 2;
        if (nt + 1 < NTILES)  // hint next B tile into cache
            __builtin_prefetch(vp + ((size_t)(bhw + 16) * G + g) * IND, 0, 0);

        v8f c = cb;
#pragma unroll
        for (int kc = 0; kc < 4; ++kc) {
            v2f bm;
            bm.x = Brow[kc * 4 + 0];
            bm.y = Brow[kc * 4 + 1];
            c = __builtin_amdgcn_wmma_f32_16x16x4_f32(
                    false, a[kc], false, bm, (short)0, c, false, false);
        }

        int b = bhw / OHW, hw = bhw % OHW;
        size_t bohw = ((size_t)b * OC + o) * OHW + hw;
        float* dst = votes + (bohw * G + g) * OD + half * 8;
#pragma unroll
        for (int r = 0; r < 8; ++r) dst[r] = c[r];
    }
}

// ---------------------------------------------------------------------------
// Kernel 3: EM stats per (b,o,h,w) site.  Stages the 288x16 votes tile into
// LDS via GLOBAL_LOAD_ASYNC_TO_LDS_B128 (ASYNCcnt path, no VGPR round-trip),
// then computes r_hat, sum_r, u, sigma, a_hat, 1/sqrt(prod(2pi*sigma)).
// ---------------------------------------------------------------------------
__global__ __launch_bounds__(512)
void stats_kernel(int first,
                  const float* __restrict__ votes, const float* __restrict__ ap,
                  const float* __restrict__ run,  const float* __restrict__ denom,
                  const float* __restrict__ beta_v, const float* __restrict__ beta_a,
                  const float* __restrict__ lamda,
                  float* __restrict__ u_out, float* __restrict__ sig_out,
                  float* __restrict__ ahat_out, float* __restrict__ invsp_out) {
    __shared__ float vt[G * OD];      // 18 KB votes tile
    __shared__ float rh[G];
    __shared__ float red[512];
    __shared__ float uvec[OD], svec[OD];
    __shared__ float ssum;

    int t = threadIdx.x;
    int bohw = blockIdx.x;
    int b = bohw / (OC * OHW);
    int rem = bohw % (OC * OHW);
    int o = rem / OHW, hw = rem % OHW;
    int h = hw / OWW, w = hw % OWW;
    int bhw = b * OHW + hw;

    // stage votes tile into LDS with async global->LDS copies (B128 each)
    {
        unsigned long long gbase =
            (unsigned long long)(const void*)(votes + (size_t)bohw * G * OD);
        unsigned ldsbase = (unsigned)(unsigned long long)(void*)vt;  // low 32 bits of the address give the LDS offset
        for (int i = t; i < (G * OD) / 4; i += 512) {
            unsigned long long ga = gbase + (unsigned long long)i * 16u;
            unsigned la = ldsbase + (unsigned)i * 16u;
            asm volatile("global_load_async_to_lds_b128 %0, %1, off"
                         :: "v"(la), "v"(ga) : "memory");
        }
    }
    // overlap: r_hat while the async copies are in flight
    if (t < G) {
        float aval = ap[(size_t)bhw * G + t];
        float r = first ? (1.0f / (float)OC)
                        : run[((size_t)b * (OC * OHW) + rem) * G + t] /
                          denom[b * G + t];
        rh[t] = r * aval;
    }
    asm volatile("s_wait_asynccnt 0x0" ::: "memory");
    __syncthreads();

    // sum_r = sum_g rh[g]
    red[t] = (t < G) ? rh[t] : 0.0f;
    __syncthreads();
    for (int s = 256; s > 0; s >>= 1) {
        if (t < s) red[t] += red[t + s];
        __syncthreads();
    }
    if (t == 0) ssum = red[0];
    __syncthreads();
    float sum_r = ssum;

    int d = t & 15, gg = t >> 4;     // 32 g-groups x 9 g each, 16 d lanes

    // u[d] = sum_g rh[g]*votes[g][d] / sum_r
    float pu = 0.0f;
#pragma unroll
    for (int k = 0; k < 9; ++k) {
        int g = gg * 9 + k;
        pu += rh[g] * vt[g * OD + d];
    }
    __syncthreads();
    red[t] = pu;
    __syncthreads();
    for (int s = 16; s > 0; s >>= 1) {
        if (gg < s) red[gg * 16 + d] += red[(gg + s) * 16 + d];
        __syncthreads();
    }
    if (t < OD) uvec[t] = red[t] / sum_r;
    __syncthreads();

    // sigma[d] = sum_g rh[g]*(v-u)^2 / sum_r
    float ud = uvec[d];
    float ps = 0.0f;
#pragma unroll
    for (int k = 0; k < 9; ++k) {
        int g = gg * 9 + k;
        float df = vt[g * OD + d] - ud;
        ps += rh[g] * df * df;
    }
    __syncthreads();
    red[t] = ps;
    __syncthreads();
    for (int s = 16; s > 0; s >>= 1) {
        if (gg < s) red[gg * 16 + d] += red[(gg + s) * 16 + d];
        __syncthreads();
    }
    if (t < OD) svec[t] = red[t] / sum_r;
    __syncthreads();

    if (t == 0) {
        float bv  = beta_v[(o * OHH + h) * OWW + w];
        float ba  = beta_a[(o * OHH + h) * OWW + w];
        float lam = lamda[0];
        float lsum = 0.0f, prod = 1.0f;
#pragma unroll
        for (int dd = 0; dd < OD; ++dd) {
            lsum += 0.5f * __logf(svec[dd]);
            prod *= 6.2832f * svec[dd];       // TWO_PI = 2*3.1416 per reference
        }
        float cost = ((float)OD * bv + lsum) * sum_r;
        float ah = 1.0f / (1.0f + __expf(-lam * (ba - cost)));
        ahat_out[bohw]  = ah;
        invsp_out[bohw] = rsqrtf(prod);
    }
    if (t < OD) {
        u_out[(size_t)bohw * OD + t]   = uvec[t];
        sig_out[(size_t)bohw * OD + t] = svec[t];
    }
}

// ---------------------------------------------------------------------------
// Kernel 4: R update.  R_un = a_hat * exp(-S * invsp)  (matches reference:
// the 1/sqrt(prod) sits inside the exp).  Normalizer via f32 global atomics.
// run layout: [b][o*OHW+hw][g]
// ---------------------------------------------------------------------------
__global__ __launch_bounds__(G)
void rupdate_kernel(const float* __restrict__ votes,
                    const float* __restrict__ u_in, const float* __restrict__ sig_in,
                    const float* __restrict__ ahat_in, const float* __restrict__ invsp_in,
                    float* __restrict__ run, float* __restrict__ denom) {
    __shared__ float uvec[OD], i2s[OD];
    __shared__ float sa, sp;
    int t = threadIdx.x;                 // g
    int bohw = blockIdx.x;
    int b = bohw / (OC * OHW);
    int rem = bohw % (OC * OHW);

    if (t < OD) {
        uvec[t] = u_in[(size_t)bohw * OD + t];
        i2s[t]  = 0.5f / sig_in[(size_t)bohw * OD + t];
    }
    if (t == 0) { sa = ahat_in[bohw]; sp = invsp_in[bohw]; }
    __syncthreads();

    const float* vrow = votes + ((size_t)bohw * G + t) * OD;
    float S = 0.0f;
#pragma unroll
    for (int d = 0; d < OD; ++d) {
        float df = vrow[d] - uvec[d];
        S += df * df * i2s[d];
    }
    float r = sa * __expf(-S * sp);
    run[((size_t)b * (OC * OHW) + rem) * G + t] = r;
    atomicAdd(&denom[b * G + t], r);
}

// ---------------------------------------------------------------------------
// Kernel 5: gather output: out[b, o*17+0, h, w]=a_hat; out[b, o*17+1+d]=u[d]
// ---------------------------------------------------------------------------
__global__ void out_kernel(const float* __restrict__ ahat,
                           const float* __restrict__ u,
                           float* __restrict__ out) {
    const int TOT = NB * OC * (OD + 1) * OHW;
    int t = blockIdx.x * blockDim.x + threadIdx.x;
    if (t >= TOT) return;
    int b   = t / (OC * (OD + 1) * OHW);
    int rem = t % (OC * (OD + 1) * OHW);
    int ch = rem / OHW, hw = rem % OHW;
    int o = ch / (OD + 1), k = ch % (OD + 1);
    size_t bohw = ((size_t)b * OC + o) * OHW + hw;
    out[t] = (k == 0) ? ahat[bohw] : u[bohw * OD + (k - 1)];
}

// ---------------------------------------------------------------------------
extern "C" void kernel_launch(void* const* d_in, const int* in_sizes, int n_in,
                              void* d_out, int out_size, void* d_ws, size_t ws_size,
                              hipStream_t stream) {
    const float* x      = (const float*)d_in[0];
    const float* Wm     = (const float*)d_in[1];
    const float* bias   = (const float*)d_in[2];
    const float* beta_v = (const float*)d_in[3];
    const float* beta_a = (const float*)d_in[4];
    const float* lamda  = (const float*)d_in[5];
    float* out = (float*)d_out;
    (void)in_sizes; (void)n_in; (void)out_size; (void)ws_size;

    char* ws = (char*)d_ws;
    size_t off = 0;
    auto alloc = [&](size_t nfl) {
        float* p = (float*)(ws + off);
        off += ((nfl * sizeof(float) + 255) / 256) * 256;
        return p;
    };
    float* vp    = alloc((size_t)NBHW * G * IND);     // 5.3 MB
    float* ap    = alloc((size_t)NBHW * G);           // 0.33 MB
    float* votes = alloc((size_t)NBOHW * G * OD);     // 170 MB (L2-resident)
    float* run   = alloc((size_t)NB * OC * OHW * G);  // 10.6 MB
    float* denom = alloc((size_t)NB * G);
    float* uu    = alloc((size_t)NBOHW * OD);
    float* sg    = alloc((size_t)NBOHW * OD);
    float* ah    = alloc((size_t)NBOHW);
    float* isp   = alloc((size_t)NBOHW);

    {
        int n = NBHW * G;
        pack_kernel<<<(n + 255) / 256, 256, 0, stream>>>(x, vp, ap);
    }
    votes_kernel<<<dim3(OC, G), 32, 0, stream>>>(Wm, bias, vp, votes);

    for (int it = 0; it < 3; ++it) {
        stats_kernel<<<NBOHW, 512, 0, stream>>>(it == 0 ? 1 : 0, votes, ap, run,
                                                denom, beta_v, beta_a, lamda,
                                                uu, sg, ah, isp);
        if (it < 2) {   // last iteration's R update is dead code
            hipMemsetAsync(denom, 0, (size_t)NB * G * sizeof(float), stream);
            rupdate_kernel<<<NBOHW, G, 0, stream>>>(votes, uu, sg, ah, isp,
                                                    run, denom);
        }
    }
    {
        const int TOT = NB * OC * (OD + 1) * OHW;
        out_kernel<<<(TOT + 255) / 256, 256, 0, stream>>>(ah, uu, out);
    }
}